// WeightedAttention_4217657885291
// MI455X (gfx1250) — compile-verified
//
#include <hip/hip_runtime.h>
#include <stdint.h>

// Weighted segment-softmax pooling for MI455X (gfx1250).
// Memory-bound: ~1.09 GB traffic -> ~47us floor @ 23.3 TB/s.
// CDNA5 features used: v_wmma_f32_16x16x4_f32 (gate GEMV),
// global_load_async_to_lds_b128 + s_wait_asynccnt (x staging),
// global_atomic_max_num_f32 / global_atomic_add_f32 (segment scatter ops).

#define DFEAT 128
#define ROWPAD 132   // LDS row stride in floats (132 % 64 = 4 -> conflict-free A reads)

typedef __attribute__((ext_vector_type(2))) float v2f;
typedef __attribute__((ext_vector_type(8))) float v8f;

__device__ __forceinline__ void atom_add_f32(float* p, float v) {
  asm volatile("global_atomic_add_f32 %0, %1, off scope:SCOPE_DEV"
               :: "v"(p), "v"(v) : "memory");
}
__device__ __forceinline__ void atom_max_f32(float* p, float v) {
  asm volatile("global_atomic_max_num_f32 %0, %1, off scope:SCOPE_DEV"
               :: "v"(p), "v"(v) : "memory");
}
__device__ __forceinline__ void async_load_b128(unsigned lds_byte_off, const void* gaddr) {
  // VDST = LDS byte offset (VGPR), VADDR = 64-bit global address (VGPR pair)
  asm volatile("global_load_async_to_lds_b128 %0, %1, off"
               :: "v"(lds_byte_off), "v"(gaddr) : "memory");
}
__device__ __forceinline__ void wait_async0() {
  asm volatile("s_wait_asynccnt 0" ::: "memory");
}

// ---------------- init: zero out/denom, gmax = -inf ----------------
__global__ void wa_init_kernel(float* __restrict__ out, float* __restrict__ gmax,
                               float* __restrict__ denom, int outElems, int S) {
  int i = blockIdx.x * blockDim.x + threadIdx.x;
  int stride = gridDim.x * blockDim.x;
  for (int j = i; j < outElems; j += stride) out[j] = 0.0f;
  for (int j = i; j < S; j += stride) { gmax[j] = -__builtin_inff(); denom[j] = 0.0f; }
}

// ---------------- pass 1: gate = x.Wg + bg, scatter-max ----------------
// 4 waves/block, 16 rows/wave (64 rows/block). Each wave stages its 16x128
// tile of x into LDS with async b128 loads, then runs 32 WMMA_F32_16x16x4_F32
// with B = Wg[k] broadcast across all N columns -> D[m][*] = dot(x[m], Wg).
__global__ void __launch_bounds__(128)
wa_gate_kernel(const float* __restrict__ x, const long long* __restrict__ idx,
               const float* __restrict__ Wg, const float* __restrict__ bgp,
               float* __restrict__ gate, float* __restrict__ gmax, int N) {
  __shared__ float sWg[DFEAT];
  __shared__ float tile[4][16 * ROWPAD];

  const int tid  = threadIdx.x;
  const int lane = tid & 31;
  const int wave = tid >> 5;

  if (tid < DFEAT) sWg[tid] = Wg[tid];
  __syncthreads();

  const float bg = bgp[0];
  const int rowBase = (blockIdx.x * 4 + wave) * 16;   // wave-uniform
  if (rowBase >= N) return;

  if (rowBase + 16 > N) {
    // scalar tail fallback (no WMMA; EXEC may diverge safely here)
    for (int r = rowBase; r < N; ++r) {
      float p = 0.0f;
      #pragma unroll
      for (int q = 0; q < 4; ++q)
        p += x[(size_t)r * DFEAT + lane * 4 + q] * sWg[lane * 4 + q];
      for (int off = 16; off > 0; off >>= 1) p += __shfl_down(p, off);
      if (lane == 0) {
        float g = p + bg;
        gate[r] = g;
        atom_max_f32(&gmax[(int)idx[r]], g);
      }
    }
    return;
  }

  float* myTile = &tile[wave][0];
  const unsigned ldsBase = (unsigned)(uintptr_t)(void*)myTile;  // low 32 bits = LDS offset
  const float* src = x + (size_t)rowBase * DFEAT;

  // Stage 16 rows: one fully-coalesced 512B async transfer per row per wave.
  #pragma unroll
  for (int r = 0; r < 16; ++r) {
    async_load_b128(ldsBase + (unsigned)(r * ROWPAD + lane * 4) * 4u,
                    src + (size_t)r * DFEAT + lane * 4);
  }
  wait_async0();

  // WMMA A layout (32-bit, 16x4): lanes 0-15 = rows, VGPR0/1 = K {0,1} (hi=0)
  // or K {2,3} (hi=1). B (4x16): VGPR0 = rows K0/K1, VGPR1 = rows K2/K3.
  const int m  = lane & 15;
  const int hi = lane >> 4;
  v8f acc = {};
  #pragma unroll
  for (int k0 = 0; k0 < DFEAT; k0 += 4) {
    v2f a, b;
    a.x = myTile[m * ROWPAD + k0 + 2 * hi];
    a.y = myTile[m * ROWPAD + k0 + 2 * hi + 1];
    b.x = sWg[k0 + hi];        // broadcast across N columns
    b.y = sWg[k0 + 2 + hi];
    acc = __builtin_amdgcn_wmma_f32_16x16x4_f32(false, a, false, b,
                                                (short)0, acc, false, false);
  }

  // Column N=0 lives in lane 0 (rows 0..7) and lane 16 (rows 8..15).
  if (m == 0) {
    const int rbase = rowBase + hi * 8;
    #pragma unroll
    for (int j = 0; j < 8; ++j) {
      float g = acc[j] + bg;
      int row = rbase + j;
      gate[row] = g;
      atom_max_f32(&gmax[(int)idx[row]], g);
    }
  }
}

// ---------------- pass 2: e = w*exp(g - gmax), scatter-add denom ----------------
__global__ void wa_softmax_kernel(const long long* __restrict__ idx,
                                  const float* __restrict__ weights,
                                  const float* __restrict__ gmax,
                                  float* __restrict__ gate,
                                  float* __restrict__ denom, int N) {
  int i = blockIdx.x * blockDim.x + threadIdx.x;
  if (i >= N) return;
  int seg = (int)idx[i];
  float e = weights[i] * __expf(gate[i] - gmax[seg]);
  gate[i] = e;                 // reuse gate[] as numerator
  atom_add_f32(&denom[seg], e);
}

// ---------------- pass 3: out[s,:] += (e/denom)*x[i,:], run-compacted ----------------
// Sorted index: each wave owns 64 consecutive rows, accumulates in registers
// per segment run, flushes 128 atomic f32 adds only at segment boundaries.
__global__ void __launch_bounds__(256)
wa_pool_kernel(const float* __restrict__ x, const long long* __restrict__ idx,
               const float* __restrict__ gate, const float* __restrict__ denom,
               float* __restrict__ out, int N) {
  const int lane = threadIdx.x & 31;
  const int waveGlobal = blockIdx.x * (blockDim.x >> 5) + (threadIdx.x >> 5);
  int r0 = waveGlobal * 64;
  if (r0 >= N) return;
  int rEnd = r0 + 64; if (rEnd > N) rEnd = N;

  const int c = lane * 4;
  float4 acc = make_float4(0.f, 0.f, 0.f, 0.f);
  int curseg = (int)idx[r0];

  for (int r = r0; r < rEnd; ++r) {
    int seg = (int)idx[r];                       // wave-uniform -> scalar load
    if (seg != curseg) {
      float* o = out + (size_t)curseg * DFEAT + c;
      atom_add_f32(o + 0, acc.x); atom_add_f32(o + 1, acc.y);
      atom_add_f32(o + 2, acc.z); atom_add_f32(o + 3, acc.w);
      acc = make_float4(0.f, 0.f, 0.f, 0.f);
      curseg = seg;
    }
    float a = gate[r] / (denom[seg] + 1e-13f);
    const float4 xv = *(const float4*)(x + (size_t)r * DFEAT + c);
    acc.x += a * xv.x; acc.y += a * xv.y;
    acc.z += a * xv.z; acc.w += a * xv.w;
  }
  float* o = out + (size_t)curseg * DFEAT + c;
  atom_add_f32(o + 0, acc.x); atom_add_f32(o + 1, acc.y);
  atom_add_f32(o + 2, acc.z); atom_add_f32(o + 3, acc.w);
}

extern "C" void kernel_launch(void* const* d_in, const int* in_sizes, int n_in,
                              void* d_out, int out_size, void* d_ws, size_t ws_size,
                              hipStream_t stream) {
  const float*     x       = (const float*)d_in[0];
  const long long* idx     = (const long long*)d_in[1];   // jnp.int64
  const float*     weights = (const float*)d_in[2];
  const float*     Wg      = (const float*)d_in[3];
  const float*     bg      = (const float*)d_in[4];
  (void)n_in; (void)ws_size;

  const int N = in_sizes[0] / DFEAT;     // 1048576
  const int S = out_size   / DFEAT;      // 65536
  float* out   = (float*)d_out;
  float* gate  = (float*)d_ws;           // N floats
  float* gmax  = gate + N;               // S floats
  float* denom = gmax + S;               // S floats   (~4.7 MB total)

  wa_init_kernel<<<1024, 256, 0, stream>>>(out, gmax, denom, out_size, S);

  const int gateBlocks = (N + 63) / 64;          // 64 rows / block (4 waves x 16)
  wa_gate_kernel<<<gateBlocks, 128, 0, stream>>>(x, idx, Wg, bg, gate, gmax, N);

  wa_softmax_kernel<<<(N + 255) / 256, 256, 0, stream>>>(idx, weights, gmax, gate, denom, N);

  const int waves = (N + 63) / 64;               // 64 rows / wave
  wa_pool_kernel<<<(waves + 7) / 8, 256, 0, stream>>>(x, idx, gate, denom, out, N);
}